// MetaSelectInput_71236327571648
// MI455X (gfx1250) — compile-verified
//
#include <hip/hip_runtime.h>

// FPN ROI-align (crop_and_resize, bilinear, extrapolation=0) for MI455X/gfx1250.
// Memory-bound: ~200MB NT stores + L2-resident feature-map gathers (~85MB, fits
// in 192MB L2). CDNA5 paths: global_load_async_to_lds_b128 double-buffered
// staging, s_wait_asynccnt pipelining, non-temporal B128 output stores.

typedef float v4f __attribute__((ext_vector_type(4)));

__device__ __forceinline__ void async_load_b128(unsigned lds_addr, const v4f* gaddr) {
    asm volatile("global_load_async_to_lds_b128 %0, %1, off"
                 :: "v"(lds_addr), "v"(gaddr)
                 : "memory");
}
__device__ __forceinline__ void wait_async_le4() {
    asm volatile("s_wait_asynccnt 0x4" ::: "memory");
}
__device__ __forceinline__ void wait_async_le0() {
    asm volatile("s_wait_asynccnt 0x0" ::: "memory");
}

__global__ __launch_bounds__(256) void roi_pyramid_kernel(
    const float* __restrict__ gt,
    const float* __restrict__ fm0, const float* __restrict__ fm1,
    const float* __restrict__ fm2, const float* __restrict__ fm3,
    const float* __restrict__ fm4,
    float* __restrict__ out, int N, int M)
{
    const int t    = threadIdx.x;
    const int tile = blockIdx.x;
    const int n    = tile / 5;          // box index 0..N-1
    const int l    = tile - n * 5;      // level 0..4
    const int b    = n / M;             // batch image index

    // Pure-ALU level parameters (no rodata lookup on the critical path).
    const int   S  = 128 >> l;                 // 128,64,32,16,8
    const float sv = (float)(8 << l);          // 8,16,32,64,128

    // Branch-free pointer select (cselect chain, no indexed stack array).
    const float* fmp = fm0;
    fmp = (l == 1) ? fm1 : fmp;
    fmp = (l == 2) ? fm2 : fmp;
    fmp = (l == 3) ? fm3 : fmp;
    fmp = (l == 4) ? fm4 : fmp;
    const v4f* __restrict__ fmv = (const v4f*)fmp;

    // Per-position tables: corner element offsets (in v4f units, b excluded),
    // blend weights, validity.
    __shared__ int   s_otl[49], s_otr[49], s_obl[49], s_obr[49], s_valid[49];
    __shared__ float s_wy[49], s_wx[49];
    __shared__ v4f   abuf[2 * 4 * 4 * 64];     // [buf][corner][pg][chan4] = 32 KB

    const float bx1 = gt[n * 5 + 0];
    const float by1 = gt[n * 5 + 1];
    const float bx2 = gt[n * 5 + 2];
    const float by2 = gt[n * 5 + 3];

    if (t < 49) {
        // Mirror reference op order: /stride /(S-1) then *(S-1).
        const int   py  = t / 7, px = t - py * 7;
        const float den = (float)S - 1.0f;
        const float ty  = (float)py / 6.0f;
        const float tx  = (float)px / 6.0f;
        const float ny1 = (by1 / sv) / den, ny2 = (by2 / sv) / den;
        const float nx1 = (bx1 / sv) / den, nx2 = (bx2 / sv) / den;
        const float ys  = (ny1 + ty * (ny2 - ny1)) * den;
        const float xs  = (nx1 + tx * (nx2 - nx1)) * den;
        const int vy = (ys >= 0.0f) && (ys <= den);
        const int vx = (xs >= 0.0f) && (xs <= den);
        const float ylo = floorf(ys), xlo = floorf(xs);
        s_wy[t] = ys - ylo;
        s_wx[t] = xs - xlo;
        const int y0  = (int)fminf(fmaxf(ylo,        0.0f), den);
        const int y1i = (int)fminf(fmaxf(ylo + 1.0f, 0.0f), den);
        const int x0  = (int)fminf(fmaxf(xlo,        0.0f), den);
        const int x1i = (int)fminf(fmaxf(xlo + 1.0f, 0.0f), den);
        s_otl[t] = (y0  * S + x0 ) * 64;
        s_otr[t] = (y0  * S + x1i) * 64;
        s_obl[t] = (y1i * S + x0 ) * 64;
        s_obr[t] = (y1i * S + x1i) * 64;
        s_valid[t] = vy && vx;
    }
    if (t == 0 && l == 0) {
        // box_ids output (second tuple element) appended after rois.
        out[(size_t)N * 62720 + n] = (float)(n / M);
    }
    __syncthreads();

    const int    chan4 = t & 63;     // float4 channel chunk (0..63) -> 256 ch
    const int    pg    = t >> 6;     // position group 0..3 (wave-uniform)
    const size_t baseB = (size_t)b * S * S * 64 + chan4;
    v4f* __restrict__ outv = (v4f*)out;
    const size_t outBase = (size_t)n * 49 * 320 + (size_t)l * 64 + chan4;

    #define LDSOFF(buf, corner) \
        ((unsigned)(unsigned long long)(&abuf[(((buf) * 4 + (corner)) * 4 + pg) * 64 + chan4]))

    auto issue = [&](int p, int buf) {
        async_load_b128(LDSOFF(buf, 0), fmv + (baseB + s_otl[p]));
        async_load_b128(LDSOFF(buf, 1), fmv + (baseB + s_otr[p]));
        async_load_b128(LDSOFF(buf, 2), fmv + (baseB + s_obl[p]));
        async_load_b128(LDSOFF(buf, 3), fmv + (baseB + s_obr[p]));
    };

    int buf = 0;
    issue(pg, 0);                                 // prologue: stage first tile
    for (int pb = 0; pb < 49; pb += 4) {
        const int p  = pb + pg;                   // wave-uniform
        const int pn = p + 4;
        const bool hasnext = (pn < 49);
        if (hasnext) issue(pn, buf ^ 1);          // overlap next DMA with blend
        if (hasnext) wait_async_le4();            // oldest 4 (current buf) done
        else         wait_async_le0();
        if (p < 49) {
            v4f tl = abuf[((buf * 4 + 0) * 4 + pg) * 64 + chan4];
            v4f tr = abuf[((buf * 4 + 1) * 4 + pg) * 64 + chan4];
            v4f bl = abuf[((buf * 4 + 2) * 4 + pg) * 64 + chan4];
            v4f br = abuf[((buf * 4 + 3) * 4 + pg) * 64 + chan4];
            const float wx = s_wx[p], wy = s_wy[p];
            v4f top = tl + (tr - tl) * wx;
            v4f bot = bl + (br - bl) * wx;
            v4f res = top + (bot - top) * wy;
            if (!s_valid[p]) { res.x = 0.0f; res.y = 0.0f; res.z = 0.0f; res.w = 0.0f; }
            __builtin_nontemporal_store(res, &outv[outBase + (size_t)p * 320]);
        }
        buf ^= 1;
    }
    #undef LDSOFF
}

extern "C" void kernel_launch(void* const* d_in, const int* in_sizes, int n_in,
                              void* d_out, int out_size, void* d_ws, size_t ws_size,
                              hipStream_t stream) {
    (void)n_in; (void)out_size; (void)d_ws; (void)ws_size;
    const float* gt  = (const float*)d_in[0];
    const float* fm0 = (const float*)d_in[1];
    const float* fm1 = (const float*)d_in[2];
    const float* fm2 = (const float*)d_in[3];
    const float* fm3 = (const float*)d_in[4];
    const float* fm4 = (const float*)d_in[5];

    const int N = in_sizes[0] / 5;                    // 800 boxes
    const int B = in_sizes[1] / (128 * 128 * 256);    // 8 images
    const int M = N / B;                              // 100 boxes/image

    dim3 grid(N * 5);   // one block per (box, level)
    dim3 block(256);    // 8 wave32s; 64 chan4 lanes x 4 position groups
    hipLaunchKernelGGL(roi_pyramid_kernel, grid, block, 0, stream,
                       gt, fm0, fm1, fm2, fm3, fm4, (float*)d_out, N, M);
}